// EnhancedGNNDetector_26018911879668
// MI455X (gfx1250) — compile-verified
//
#include <hip/hip_runtime.h>
#include <math.h>

#define NN 50000
#define EE 800000
#define GG 64
#define DIN 128
#define DH 256
#define DH2 128
#define ATTD 128
#define NHEAD 4
#define CHEAD 32
#define EPSV 1e-5f

typedef float v2f __attribute__((ext_vector_type(2)));
typedef float v8f __attribute__((ext_vector_type(8)));

// ---------------- utility ----------------
__global__ void k_zero(float* p, long long n) {
  long long i = (long long)blockIdx.x * blockDim.x + threadIdx.x;
  if (i < n) p[i] = 0.f;
}
__global__ void k_fill(float* p, long long n, float v) {
  long long i = (long long)blockIdx.x * blockDim.x + threadIdx.x;
  if (i < n) p[i] = v;
}
__global__ void k_add(float* a, const float* __restrict__ b, long long n) {
  long long i = (long long)blockIdx.x * blockDim.x + threadIdx.x;
  if (i < n) a[i] += b[i];
}

// ---------------- degree / counts ----------------
__global__ void k_deg(const int* __restrict__ ei, float* deg) {
  int e = blockIdx.x * blockDim.x + threadIdx.x;
  if (e < EE) atomicAdd(&deg[ei[EE + e]], 1.f);
}
__global__ void k_dinv(const float* __restrict__ deg, float* dinv) {
  int i = blockIdx.x * blockDim.x + threadIdx.x;
  if (i < NN) { float d = deg[i]; dinv[i] = d > 0.f ? rsqrtf(d) : 0.f; }
}
__global__ void k_counts(const int* __restrict__ batch, float* counts) {
  int i = blockIdx.x * blockDim.x + threadIdx.x;
  if (i < NN) atomicAdd(&counts[batch[i]], 1.f);
}
__global__ void k_cinv(const float* __restrict__ counts, float* cinv) {
  int g = threadIdx.x;
  if (g < GG) cinv[g] = 1.f / fmaxf(counts[g], 1.f);
}

// ---------------- exact-f32 WMMA GEMM: C[n x k] = A[n x m] @ W[m x k] ----------------
// One wave computes one 16x16 output tile via V_WMMA_F32_16X16X4_F32.
// A-frag (16x4 f32): lanes 0-15 -> K=0,1 ; lanes 16-31 -> K=2,3 (ISA 7.12.2).
// C/D layout: VGPR v holds row (lane<16 ? v : v+8), col = lane&15.
__global__ void k_wmma_gemm(const float* __restrict__ A, const float* __restrict__ W,
                            float* __restrict__ C, int n, int m, int k) {
  int lane  = threadIdx.x;                       // 0..31
  int tileM = blockIdx.x;                        // n/16 tiles (n divisible by 16)
  int tileN = blockIdx.y * blockDim.y + threadIdx.y;
  if (tileN * 16 >= k) return;                   // wave-uniform exit
  int row = lane & 15;
  int kh  = (lane >> 4) << 1;                    // 0 or 2
  const float* arow = A + (long long)(tileM * 16 + row) * m;
  const float* bcol = W + (long long)(tileN * 16 + row);
  v8f acc = {};
  for (int kk = 0; kk < m; kk += 4) {
    v2f a; a.x = arow[kk + kh];                 a.y = arow[kk + kh + 1];
    v2f b; b.x = bcol[(long long)(kk + kh) * k]; b.y = bcol[(long long)(kk + kh + 1) * k];
    acc = __builtin_amdgcn_wmma_f32_16x16x4_f32(false, a, false, b,
                                                (short)0, acc, false, false);
  }
  int r0 = tileM * 16 + (lane >> 4) * 8;
  long long cc = (long long)(tileN * 16) + (lane & 15);
  for (int v = 0; v < 8; ++v)
    C[(long long)(r0 + v) * k + cc] = acc[v];
}

// ---------------- GCN scatter (real edges) + finish (self loop, bias, relu) ----------------
__global__ void k_gcn_scatter(const int* __restrict__ ei, const float* __restrict__ t,
                              const float* __restrict__ dinv, float* agg, int D) {
  int e = blockIdx.x;        // 0..EE-1
  int d = threadIdx.x;       // blockDim.x == D
  int s = ei[e], dd = ei[EE + e];
  float w = dinv[s] * dinv[dd];
  atomicAdd(&agg[(long long)dd * D + d], t[(long long)s * D + d] * w);
}
__global__ void k_gcn_finish(const float* __restrict__ t, const float* __restrict__ dinv,
                             const float* __restrict__ bias, float* h, int D) {
  int n = blockIdx.x, d = threadIdx.x;
  float di = dinv[n];
  long long idx = (long long)n * D + d;
  float v = h[idx] + t[idx] * di * di + bias[d];
  h[idx] = fmaxf(v, 0.f);
}

// ---------------- GraphNorm ----------------
__global__ void k_gn_mean(const float* __restrict__ h, const int* __restrict__ batch,
                          float* mean, int D) {
  int n = blockIdx.x, d = threadIdx.x;
  atomicAdd(&mean[(long long)batch[n] * D + d], h[(long long)n * D + d]);
}
__global__ void k_gn_center(float* h, const int* __restrict__ batch,
                            const float* __restrict__ mean, const float* __restrict__ cinv,
                            const float* __restrict__ ms, float* var, int D) {
  int n = blockIdx.x, d = threadIdx.x;
  int g = batch[n];
  long long idx = (long long)n * D + d;
  float out = h[idx] - mean[(long long)g * D + d] * cinv[g] * ms[d];
  h[idx] = out;
  atomicAdd(&var[(long long)g * D + d], out * out);
}
__global__ void k_gn_final(float* h, const int* __restrict__ batch,
                           const float* __restrict__ var, const float* __restrict__ cinv,
                           const float* __restrict__ w, const float* __restrict__ b, int D) {
  int n = blockIdx.x, d = threadIdx.x;
  int g = batch[n];
  long long idx = (long long)n * D + d;
  float vv = var[(long long)g * D + d] * cinv[g];
  h[idx] = w[d] * h[idx] * rsqrtf(vv + EPSV) + b[d];
}

// ---------------- GATv2 ----------------
__device__ inline void atomicMaxF(float* addr, float val) {
  int* ai = (int*)addr;
  int old = __float_as_int(*addr);
  while (__int_as_float(old) < val) {
    int assumed = old;
    old = atomicCAS(ai, assumed, __float_as_int(val));
    if (old == assumed) break;
  }
}

__global__ void k_gat_self(const float* __restrict__ xl, const float* __restrict__ xr,
                           const float* __restrict__ att, float* alphaS, float* amax) {
  int i = blockIdx.x * blockDim.x + threadIdx.x;  // N*NHEAD
  if (i >= NN * NHEAD) return;
  int n = i >> 2, h = i & 3;
  float s = 0.f;
  for (int c = 0; c < CHEAD; ++c) {
    long long idx = (long long)n * ATTD + h * CHEAD + c;
    float v = xl[idx] + xr[idx];
    v = v > 0.f ? v : 0.2f * v;
    s += v * att[h * CHEAD + c];
  }
  alphaS[i] = s;
  amax[i] = s;  // self-edge always present -> seeds segment max
}
__global__ void k_gat_edge_alpha(const int* __restrict__ ei, const float* __restrict__ xl,
                                 const float* __restrict__ xr, const float* __restrict__ att,
                                 float* alphaE, float* amax) {
  long long i = (long long)blockIdx.x * blockDim.x + threadIdx.x;  // EE*NHEAD
  if (i >= (long long)EE * NHEAD) return;
  int e = (int)(i >> 2), h = (int)(i & 3);
  int s = ei[e], d = ei[EE + e];
  float a = 0.f;
  for (int c = 0; c < CHEAD; ++c) {
    float v = xl[(long long)s * ATTD + h * CHEAD + c] + xr[(long long)d * ATTD + h * CHEAD + c];
    v = v > 0.f ? v : 0.2f * v;
    a += v * att[h * CHEAD + c];
  }
  alphaE[i] = a;
  atomicMaxF(&amax[(long long)d * NHEAD + h], a);
}
__global__ void k_gat_exp_edge(const int* __restrict__ ei, float* alphaE,
                               const float* __restrict__ amax, float* denom) {
  long long i = (long long)blockIdx.x * blockDim.x + threadIdx.x;
  if (i >= (long long)EE * NHEAD) return;
  int e = (int)(i >> 2), h = (int)(i & 3);
  int d = ei[EE + e];
  float w = expf(alphaE[i] - amax[(long long)d * NHEAD + h]);
  alphaE[i] = w;
  atomicAdd(&denom[(long long)d * NHEAD + h], w);
}
__global__ void k_gat_exp_self(float* alphaS, const float* __restrict__ amax, float* denom) {
  int i = blockIdx.x * blockDim.x + threadIdx.x;
  if (i >= NN * NHEAD) return;
  float w = expf(alphaS[i] - amax[i]);
  alphaS[i] = w;
  atomicAdd(&denom[i], w);
}
__global__ void k_gat_aggregate(const int* __restrict__ ei, const float* __restrict__ alphaE,
                                const float* __restrict__ denom, const float* __restrict__ xl,
                                float* hatt) {
  int e = blockIdx.x;          // 0..EE-1
  int hc = threadIdx.x;        // 0..127
  int h = hc >> 5;
  int s = ei[e], d = ei[EE + e];
  float coef = alphaE[(long long)e * NHEAD + h] / (denom[(long long)d * NHEAD + h] + 1e-16f);
  atomicAdd(&hatt[(long long)d * ATTD + hc], coef * xl[(long long)s * ATTD + hc]);
}
__global__ void k_gat_finish(const float* __restrict__ alphaS, const float* __restrict__ denom,
                             const float* __restrict__ xl, const float* __restrict__ att_b,
                             float* hatt) {
  int n = blockIdx.x, hc = threadIdx.x;
  int h = hc >> 5;
  float coef = alphaS[n * NHEAD + h] / (denom[n * NHEAD + h] + 1e-16f);
  long long idx = (long long)n * ATTD + hc;
  float v = hatt[idx] + coef * xl[idx] + att_b[hc];
  hatt[idx] = fmaxf(v, 0.f);
}

// ---------------- pooling + classifier ----------------
__global__ void k_pool(const float* __restrict__ hatt, const int* __restrict__ batch,
                       float* pool) {
  int n = blockIdx.x, d = threadIdx.x;  // 128
  atomicAdd(&pool[(long long)batch[n] * ATTD + d], hatt[(long long)n * ATTD + d]);
}
__global__ void k_fc1(const float* __restrict__ pool, const float* __restrict__ cinv,
                      const float* __restrict__ W, const float* __restrict__ b, float* z) {
  int g = blockIdx.x, j = threadIdx.x;  // 64
  float ci = cinv[g], s = 0.f;
  for (int d = 0; d < 128; ++d) s += pool[g * 128 + d] * ci * W[d * 64 + j];
  z[g * 64 + j] = fmaxf(s + b[j], 0.f);
}
__global__ void k_fc2(const float* __restrict__ z1, const float* __restrict__ W,
                      const float* __restrict__ b, float* z) {
  int g = blockIdx.x, j = threadIdx.x;  // 32
  float s = 0.f;
  for (int d = 0; d < 64; ++d) s += z1[g * 64 + d] * W[d * 32 + j];
  z[g * 32 + j] = fmaxf(s + b[j], 0.f);
}
__global__ void k_fc3(const float* __restrict__ z2, const float* __restrict__ W,
                      const float* __restrict__ b, float* out) {
  int g = blockIdx.x, j = threadIdx.x;  // 2
  float s = 0.f;
  for (int d = 0; d < 32; ++d) s += z2[g * 32 + d] * W[d * 2 + j];
  out[g * 2 + j] = s + b[j];
}

// ---------------- host ----------------
extern "C" void kernel_launch(void* const* d_in, const int* in_sizes, int n_in,
                              void* d_out, int out_size, void* d_ws, size_t ws_size,
                              hipStream_t stream) {
  const float* x     = (const float*)d_in[0];
  const int*   ei    = (const int*)d_in[1];
  const int*   batch = (const int*)d_in[2];
  const float* W1 = (const float*)d_in[3];  const float* b1 = (const float*)d_in[4];
  const float* W2 = (const float*)d_in[5];  const float* b2 = (const float*)d_in[6];
  const float* W3 = (const float*)d_in[7];  const float* b3 = (const float*)d_in[8];
  const float* gn1w = (const float*)d_in[9];  const float* gn1b = (const float*)d_in[10]; const float* gn1m = (const float*)d_in[11];
  const float* gn2w = (const float*)d_in[12]; const float* gn2b = (const float*)d_in[13]; const float* gn2m = (const float*)d_in[14];
  const float* gn3w = (const float*)d_in[15]; const float* gn3b = (const float*)d_in[16]; const float* gn3m = (const float*)d_in[17];
  const float* gn4w = (const float*)d_in[18]; const float* gn4b = (const float*)d_in[19]; const float* gn4m = (const float*)d_in[20];
  const float* Wl = (const float*)d_in[21]; const float* Wr = (const float*)d_in[22];
  const float* att = (const float*)d_in[23]; const float* att_b = (const float*)d_in[24];
  const float* Wc1 = (const float*)d_in[25]; const float* bc1 = (const float*)d_in[26];
  const float* Wc2 = (const float*)d_in[27]; const float* bc2 = (const float*)d_in[28];
  const float* Wc3 = (const float*)d_in[29]; const float* bc3 = (const float*)d_in[30];
  float* out = (float*)d_out;

  // workspace layout (floats) -- everything L2-resident (~170MB < 192MB L2)
  float* ws = (float*)d_ws;
  size_t o = 0;
  float* deg    = ws + o; o += NN;
  float* dinv   = ws + o; o += NN;
  float* counts = ws + o; o += GG;
  float* cinv   = ws + o; o += GG;
  float* P0 = ws + o; o += (size_t)NN * DH;
  float* P1 = ws + o; o += (size_t)NN * DH;
  float* P2 = ws + o; o += (size_t)NN * DH;
  float* alphaE = ws + o; o += (size_t)EE * NHEAD;
  float* alphaS = ws + o; o += (size_t)NN * NHEAD;
  float* amax   = ws + o; o += (size_t)NN * NHEAD;
  float* denom  = ws + o; o += (size_t)NN * NHEAD;
  float* mean   = ws + o; o += (size_t)GG * DH;
  float* var    = ws + o; o += (size_t)GG * DH;
  float* pool   = ws + o; o += (size_t)GG * ATTD;
  float* z1     = ws + o; o += (size_t)GG * 64;
  float* z2     = ws + o; o += (size_t)GG * 32;

  auto zero = [&](float* p, long long n) {
    k_zero<<<(unsigned)((n + 255) / 256), 256, 0, stream>>>(p, n);
  };
  auto gemm = [&](const float* A, const float* W, float* C, int m, int k) {
    dim3 grid(NN / 16, (unsigned)(k / 64));
    dim3 block(32, 4);
    k_wmma_gemm<<<grid, block, 0, stream>>>(A, W, C, NN, m, k);
  };
  auto gnorm = [&](float* h, int D, const float* w, const float* b, const float* ms) {
    zero(mean, (long long)GG * D);
    zero(var,  (long long)GG * D);
    k_gn_mean  <<<NN, D, 0, stream>>>(h, batch, mean, D);
    k_gn_center<<<NN, D, 0, stream>>>(h, batch, mean, cinv, ms, var, D);
    k_gn_final <<<NN, D, 0, stream>>>(h, batch, var, cinv, w, b, D);
  };

  // degrees / counts
  k_fill<<<(NN + 255) / 256, 256, 0, stream>>>(deg, NN, 1.f);  // self loop
  zero(counts, GG);
  k_deg   <<<(EE + 255) / 256, 256, 0, stream>>>(ei, deg);
  k_dinv  <<<(NN + 255) / 256, 256, 0, stream>>>(deg, dinv);
  k_counts<<<(NN + 255) / 256, 256, 0, stream>>>(batch, counts);
  k_cinv  <<<1, 64, 0, stream>>>(counts, cinv);

  // GCN layer 1: x[N,128] -> h1 = P2[N,256]
  gemm(x, W1, P1, DIN, DH);
  zero(P2, (long long)NN * DH);
  k_gcn_scatter<<<EE, DH, 0, stream>>>(ei, P1, dinv, P2, DH);
  k_gcn_finish <<<NN, DH, 0, stream>>>(P1, dinv, b1, P2, DH);
  gnorm(P2, DH, gn1w, gn1b, gn1m);                 // h1 = P2

  // GCN layer 2: h1 -> h2 = P1[N,256] (with residual +h1)
  gemm(P2, W2, P0, DH, DH);
  zero(P1, (long long)NN * DH);
  k_gcn_scatter<<<EE, DH, 0, stream>>>(ei, P0, dinv, P1, DH);
  k_gcn_finish <<<NN, DH, 0, stream>>>(P0, dinv, b2, P1, DH);
  gnorm(P1, DH, gn2w, gn2b, gn2m);
  k_add<<<(unsigned)(((long long)NN * DH + 255) / 256), 256, 0, stream>>>(P1, P2, (long long)NN * DH);

  // GCN layer 3: h2 -> h3 = P2[N,128]
  gemm(P1, W3, P0, DH, DH2);
  zero(P2, (long long)NN * DH2);
  k_gcn_scatter<<<EE, DH2, 0, stream>>>(ei, P0, dinv, P2, DH2);
  k_gcn_finish <<<NN, DH2, 0, stream>>>(P0, dinv, b3, P2, DH2);
  gnorm(P2, DH2, gn3w, gn3b, gn3m);                // h3 = P2

  // GATv2: xl,xr in P0 (two N x 128 halves); hatt = P1
  float* xl = P0;
  float* xr = P0 + (size_t)NN * ATTD;
  gemm(P2, Wl, xl, DH2, ATTD);
  gemm(P2, Wr, xr, DH2, ATTD);
  k_gat_self<<<(NN * NHEAD + 255) / 256, 256, 0, stream>>>(xl, xr, att, alphaS, amax);
  k_gat_edge_alpha<<<(unsigned)(((long long)EE * NHEAD + 255) / 256), 256, 0, stream>>>(
      ei, xl, xr, att, alphaE, amax);
  zero(denom, (long long)NN * NHEAD);
  k_gat_exp_edge<<<(unsigned)(((long long)EE * NHEAD + 255) / 256), 256, 0, stream>>>(
      ei, alphaE, amax, denom);
  k_gat_exp_self<<<(NN * NHEAD + 255) / 256, 256, 0, stream>>>(alphaS, amax, denom);
  float* hatt = P1;
  zero(hatt, (long long)NN * ATTD);
  k_gat_aggregate<<<EE, ATTD, 0, stream>>>(ei, alphaE, denom, xl, hatt);
  k_gat_finish   <<<NN, ATTD, 0, stream>>>(alphaS, denom, xl, att_b, hatt);
  gnorm(hatt, ATTD, gn4w, gn4b, gn4m);

  // pool + classifier
  zero(pool, (long long)GG * ATTD);
  k_pool<<<NN, ATTD, 0, stream>>>(hatt, batch, pool);
  k_fc1<<<GG, 64, 0, stream>>>(pool, cinv, Wc1, bc1, z1);
  k_fc2<<<GG, 32, 0, stream>>>(z1, Wc2, bc2, z2);
  k_fc3<<<GG, 2, 0, stream>>>(z2, Wc3, bc3, out);

  (void)in_sizes; (void)n_in; (void)out_size; (void)ws_size;
}